// JastrowFactor_72971494359230
// MI455X (gfx1250) — compile-verified
//
#include <hip/hip_runtime.h>

typedef __attribute__((ext_vector_type(2))) float v2f;
typedef __attribute__((ext_vector_type(8))) float v8f;

#define WAVES_PER_BLOCK 8
#define N_EL 64

__global__ __launch_bounds__(256, 1)
void jastrow_wmma_kernel(const float* __restrict__ ecoord,
                         const float* __restrict__ alpha_par,
                         const float* __restrict__ alpha_anti,
                         const int* __restrict__ n_up_ptr,
                         float* __restrict__ out,
                         int nbatch)
{
    // Per-wave staging: 192 coords + 64 squared norms = 1 KB/wave, 8 KB/block.
    __shared__ float lX[WAVES_PER_BLOCK][N_EL * 3];
    __shared__ float lN[WAVES_PER_BLOCK][N_EL];

    const int lane = threadIdx.x & 31;
    const int wave = threadIdx.x >> 5;
    int batch = blockIdx.x * WAVES_PER_BLOCK + wave;
    if (batch >= nbatch) batch = nbatch - 1;  // wave-uniform clamp; duplicate work is benign

    const float* xb = ecoord + (size_t)batch * (N_EL * 3);

    // Stage this batch's 192 floats into LDS as 48 float4s.
    const float4* src4 = reinterpret_cast<const float4*>(xb);
    float4* dst4 = reinterpret_cast<float4*>(&lX[wave][0]);
    for (int k = lane; k < (N_EL * 3) / 4; k += 32)
        dst4[k] = src4[k];
    __syncthreads();

    // Squared norms |x_r|^2 for all 64 electrons.
    for (int r = lane; r < N_EL; r += 32) {
        float x0 = lX[wave][r * 3 + 0];
        float x1 = lX[wave][r * 3 + 1];
        float x2 = lX[wave][r * 3 + 2];
        lN[wave][r] = x0 * x0 + x1 * x1 + x2 * x2;
    }
    __syncthreads();

    // Uniform scalars.
    const float ap  = alpha_par[0];
    const float aa  = alpha_anti[0];
    const int   nup = n_up_ptr[0];
    const float cpar  = -0.25f * ap * ap;
    const float canti = -0.5f  * aa * aa;

    // Build the four 16x4 (K=3, zero-padded to 4) WMMA operands, one per row-block.
    // f32 A layout: lanes 0-15 hold {K=0,K=1}, lanes 16-31 hold {K=2,K=3}; the
    // symmetric B (= rows^T) layout reduces to the identical per-lane values.
    const bool lo   = lane < 16;
    const int  mrow = lane & 15;
    v2f op[4];
    #pragma unroll
    for (int b = 0; b < 4; ++b) {
        int r = b * 16 + mrow;
        float e0 = lX[wave][r * 3 + (lo ? 0 : 2)];
        float e1 = lo ? lX[wave][r * 3 + 1] : 0.0f;
        op[b] = (v2f){e0, e1};
    }

    float acc = 0.0f;
    const int hi8 = lo ? 0 : 8;

    // 10 tiles of the upper-triangular region of the 64x64 pair matrix.
    #pragma unroll
    for (int ti = 0; ti < 4; ++ti) {
        #pragma unroll
        for (int tj = 0; tj < 4; ++tj) {
            if (tj < ti) continue;

            v8f c = {};
            // D = A x B + 0 : Gram tile G[M][N] = x_{ti*16+M} . x_{tj*16+N}
            c = __builtin_amdgcn_wmma_f32_16x16x4_f32(
                    /*neg_a=*/false, op[ti],
                    /*neg_b=*/false, op[tj],
                    /*c_mod=*/(short)0, c,
                    /*reuse_a=*/false, /*reuse_b=*/false);

            const int   col  = tj * 16 + mrow;           // N index for this lane
            const float ncol = lN[wave][col];
            const int   rowbase = ti * 16 + hi8;         // M = rowbase + v, v in 0..7
            const float4 nr0 = *reinterpret_cast<const float4*>(&lN[wave][rowbase]);
            const float4 nr1 = *reinterpret_cast<const float4*>(&lN[wave][rowbase + 4]);
            const float nrow[8] = {nr0.x, nr0.y, nr0.z, nr0.w,
                                   nr1.x, nr1.y, nr1.z, nr1.w};

            #pragma unroll
            for (int v = 0; v < 8; ++v) {
                const int row = rowbase + v;
                float r2 = nrow[v] + ncol - 2.0f * c[v];
                r2 = fmaxf(r2, 0.0f);
                float r = __builtin_amdgcn_sqrtf(r2);
                r = fmaxf(r, 1e-10f);
                const bool  same = (row < nup) == (col < nup);
                const float asel = same ? ap   : aa;
                const float csel = same ? cpar : canti;
                const float t = csel * __builtin_amdgcn_rcpf(asel + r);
                acc += (row < col) ? t : 0.0f;   // strict upper triangle
            }
        }
    }

    // wave32 reduction.
    #pragma unroll
    for (int off = 16; off > 0; off >>= 1)
        acc += __shfl_xor(acc, off, 32);

    if (lane == 0) out[batch] = acc;
}

extern "C" void kernel_launch(void* const* d_in, const int* in_sizes, int n_in,
                              void* d_out, int out_size, void* d_ws, size_t ws_size,
                              hipStream_t stream) {
    // setup_inputs order: electron_coords, atom_coords (unused), alpha_par,
    // alpha_anti, n_electrons_up.
    const float* ecoord = (const float*)d_in[0];
    const float* ap     = (const float*)d_in[2];
    const float* aa     = (const float*)d_in[3];
    const int*   nup    = (const int*)d_in[4];
    float*       out    = (float*)d_out;

    const int nbatch = in_sizes[0] / (N_EL * 3);
    const int blocks = (nbatch + WAVES_PER_BLOCK - 1) / WAVES_PER_BLOCK;
    jastrow_wmma_kernel<<<blocks, 256, 0, stream>>>(ecoord, ap, aa, nup, out, nbatch);
}